// Voxelization_37915971289632
// MI455X (gfx1250) — compile-verified
//
#include <hip/hip_runtime.h>
#include <hip/hip_bf16.h>
#include <stdint.h>

// ---------------------------------------------------------------------------
// Hard voxelization (KITTI SECOND config), sort-free CDNA5 implementation.
//
// Grid: 1408 x 1600 x 40 ; key = c0*64000 + (c1*40 + c2)  (lexicographic)
// Outputs (concatenated in d_out, 480001 elements):
//   voxels  float32 [20000,5,4]   @ 0
//   coors   int32   [20000,3]     @ 400000
//   npts    int32   [20000]       @ 460000
//   vnum    int32   [1]           @ 480000
// ---------------------------------------------------------------------------

#define GX 1408
#define GY 1600
#define GZ 40
#define NSLAB 1408
#define SLABVOX (GY * GZ)   // 64000 voxels per x-slab
#define NWORDS 2000         // 64000 / 32 bitmap words
#define MAXV 20000
#define MAXP 5

#if defined(__has_builtin)
#if __has_builtin(__builtin_amdgcn_global_load_async_to_lds_b32)
#define HAVE_ASYNC_LDS 1
#endif
#endif

typedef __attribute__((address_space(1))) int as1_int;
typedef __attribute__((address_space(3))) int as3_int;

__device__ __forceinline__ bool cell_of(float x, float y, float z,
                                        int& c0, int& c1, int& c2) {
  // Matches reference: valid = all(p >= mn & p < mx); coords = clip(floor((p-mn)/vs))
  bool valid = (x >= 0.0f) & (x < 70.4f) &
               (y >= -40.0f) & (y < 40.0f) &
               (z >= -3.0f) & (z < 1.0f);
  if (!valid) return false;
  c0 = (int)floorf(x / 0.05f);
  c1 = (int)floorf((y + 40.0f) / 0.05f);
  c2 = (int)floorf((z + 3.0f) / 0.1f);
  c0 = min(max(c0, 0), GX - 1);
  c1 = min(max(c1, 0), GY - 1);
  c2 = min(max(c2, 0), GZ - 1);
  return true;
}

// ---------------------------------------------------------------------------
// K0: deterministic init of outputs + workspace counters (harness poisons once)
// ---------------------------------------------------------------------------
__global__ __launch_bounds__(256) void k_init(unsigned* __restrict__ dout, int outN,
                                              unsigned* __restrict__ cand,
                                              unsigned* __restrict__ cntV,
                                              unsigned* __restrict__ keyOfVid,
                                              unsigned* __restrict__ slabCnt,
                                              unsigned* __restrict__ slabFill,
                                              unsigned* __restrict__ slabOcc) {
  int i = blockIdx.x * 256 + threadIdx.x;
  if (i < outN) dout[i] = 0u;
  if (i < MAXV * MAXP) cand[i] = 0xFFFFFFFFu;
  if (i < MAXV) { cntV[i] = 0u; keyOfVid[i] = 0u; }
  if (i < NSLAB) { slabCnt[i] = 0u; slabFill[i] = 0u; slabOcc[i] = 0u; }
}

// ---------------------------------------------------------------------------
// K1: points-per-slab histogram, LDS-aggregated (ds atomics) to cut global
//     atomic contention 4096x per block. float4 loads -> global_load_b128.
// ---------------------------------------------------------------------------
__global__ __launch_bounds__(256) void k_count(const float4* __restrict__ pts4, int N,
                                               unsigned* __restrict__ slabCnt) {
  __shared__ unsigned h[NSLAB];
  for (int i = threadIdx.x; i < NSLAB; i += 256) h[i] = 0u;
  __syncthreads();
  int base = blockIdx.x * 4096;
  for (int i = threadIdx.x; i < 4096; i += 256) {
    int p = base + i;
    if (p >= N) break;
    float4 pt = pts4[p];
    int c0, c1, c2;
    if (cell_of(pt.x, pt.y, pt.z, c0, c1, c2)) atomicAdd(&h[c0], 1u);
  }
  __syncthreads();
  for (int i = threadIdx.x; i < NSLAB; i += 256) {
    unsigned v = h[i];
    if (v) atomicAdd(&slabCnt[i], v);
  }
}

// ---------------------------------------------------------------------------
// K2/K5: Blelloch exclusive scan over 1408 counters (one 1024-thread block,
//        2048-entry LDS tree). Optionally emits total + capped voxel_num.
// ---------------------------------------------------------------------------
__global__ __launch_bounds__(1024) void k_exscan(const unsigned* __restrict__ in,
                                                 unsigned* __restrict__ out, int n,
                                                 unsigned* __restrict__ total,
                                                 int* __restrict__ vnumOut) {
  __shared__ unsigned s[2048];
  int t = threadIdx.x;
  for (int i = t; i < 2048; i += 1024) s[i] = (i < n) ? in[i] : 0u;
  __syncthreads();
  for (int d = 1; d < 2048; d <<= 1) {
    int idx = (t + 1) * (d << 1) - 1;
    if (idx < 2048) s[idx] += s[idx - d];
    __syncthreads();
  }
  if (t == 0) {
    unsigned tot = s[2047];
    if (total) *total = tot;
    if (vnumOut) *vnumOut = (int)(tot < (unsigned)MAXV ? tot : (unsigned)MAXV);
    s[2047] = 0u;
  }
  __syncthreads();
  for (int d = 1024; d > 0; d >>= 1) {
    int idx = (t + 1) * (d << 1) - 1;
    if (idx < 2048) {
      unsigned tmp = s[idx - d];
      s[idx - d] = s[idx];
      s[idx] += tmp;
    }
    __syncthreads();
  }
  for (int i = t; i < n; i += 1024) out[i] = s[i];
}

// ---------------------------------------------------------------------------
// K3: scatter packed records {origIdx<<32 | localkey} into slab-contiguous
//     region. Order within a slab is irrelevant: stability is recovered later
//     by min-index selection. float4 load -> global_load_b128.
// ---------------------------------------------------------------------------
__global__ __launch_bounds__(256) void k_scatter(const float4* __restrict__ pts4, int N,
                                                 const unsigned* __restrict__ slabPtBase,
                                                 unsigned* __restrict__ slabFill,
                                                 unsigned long long* __restrict__ recs) {
  int p = blockIdx.x * 256 + threadIdx.x;
  if (p >= N) return;
  float4 pt = pts4[p];
  int c0, c1, c2;
  if (!cell_of(pt.x, pt.y, pt.z, c0, c1, c2)) return;
  unsigned lk = (unsigned)(c1 * GZ + c2);   // < 64000, fits in 16 bits
  unsigned pos = atomicAdd(&slabFill[c0], 1u);
  recs[(size_t)slabPtBase[c0] + pos] = ((unsigned long long)(unsigned)p << 32) | (unsigned long long)lk;
}

// ---------------------------------------------------------------------------
// K4: per-slab 64000-bit occupancy bitmap in LDS (ds atomic OR), popcount to
//     slabOcc, spill bitmap to global for K6.
// ---------------------------------------------------------------------------
__global__ __launch_bounds__(256) void k_bitmap(const unsigned long long* __restrict__ recs,
                                                const unsigned* __restrict__ ptBase,
                                                const unsigned* __restrict__ ptCnt,
                                                unsigned* __restrict__ bitmaps,
                                                unsigned* __restrict__ slabOcc) {
  __shared__ unsigned bm[NWORDS];
  __shared__ unsigned occ;
  int slab = blockIdx.x;
  for (int i = threadIdx.x; i < NWORDS; i += 256) bm[i] = 0u;
  if (threadIdx.x == 0) occ = 0u;
  __syncthreads();
  unsigned b = ptBase[slab], n = ptCnt[slab];
  for (unsigned i = threadIdx.x; i < n; i += 256) {
    __builtin_prefetch(&recs[(size_t)b + i + 256], 0, 0);  // global_prefetch_b8
    unsigned lk = (unsigned)(recs[(size_t)b + i] & 0xFFFFu);
    atomicOr(&bm[lk >> 5], 1u << (lk & 31u));
  }
  __syncthreads();
  unsigned part = 0u;
  for (int i = threadIdx.x; i < NWORDS; i += 256) {
    unsigned w = bm[i];
    bitmaps[(size_t)slab * NWORDS + i] = w;
    part += (unsigned)__popc(w);
  }
  if (part) atomicAdd(&occ, part);
  __syncthreads();
  if (threadIdx.x == 0) slabOcc[slab] = occ;
}

// ---------------------------------------------------------------------------
// K6: per-slab voxel-id assignment (prefix popcount over bitmap) + insert each
//     point's original index into its voxel's 5 sorted min-slots (atomicMin
//     cascade == "first 5 points in original order").
// ---------------------------------------------------------------------------
__global__ __launch_bounds__(256) void k_assign(const unsigned long long* __restrict__ recs,
                                                const unsigned* __restrict__ ptBase,
                                                const unsigned* __restrict__ ptCnt,
                                                unsigned* __restrict__ bitmaps,
                                                const unsigned* __restrict__ voxBase,
                                                unsigned* __restrict__ keyOfVid,
                                                unsigned* __restrict__ cntV,
                                                unsigned* __restrict__ cand) {
  __shared__ unsigned bm[NWORDS];
  __shared__ unsigned wp[NWORDS + 1];
  int slab = blockIdx.x;
  unsigned vb = voxBase[slab];
  if (vb >= (unsigned)MAXV) return;  // whole slab beyond cap -> uniform early out

  // Stage bitmap global -> LDS via async DMA (ASYNCcnt path), fallback: loads.
#if defined(HAVE_ASYNC_LDS)
  {
    as1_int* gsrc = (as1_int*)(bitmaps + (size_t)slab * NWORDS);
    as3_int* ldst = (as3_int*)(&bm[0]);
    for (int i = threadIdx.x; i < NWORDS; i += 256)
      __builtin_amdgcn_global_load_async_to_lds_b32(gsrc + i, ldst + i, 0, 0);
#if __has_builtin(__builtin_amdgcn_s_wait_asynccnt)
    __builtin_amdgcn_s_wait_asynccnt(0);
#else
    asm volatile("s_wait_asynccnt 0" ::: "memory");
#endif
  }
#else
  for (int i = threadIdx.x; i < NWORDS; i += 256)
    bm[i] = bitmaps[(size_t)slab * NWORDS + i];
#endif
  __syncthreads();

  if (threadIdx.x == 0) {  // word-prefix popcounts (2000 serial LDS adds; tiny)
    unsigned acc = 0u;
    for (int i = 0; i < NWORDS; ++i) { wp[i] = acc; acc += (unsigned)__popc(bm[i]); }
    wp[NWORDS] = acc;
  }
  __syncthreads();

  // (a) record the voxel key of each new sorted-unique voxel (for coors)
  for (int w = threadIdx.x; w < NWORDS; w += 256) {
    unsigned bits = bm[w];
    unsigned r = wp[w];
    while (bits) {
      int bit = __ffs(bits) - 1;
      bits &= bits - 1u;
      unsigned vid = vb + r;
      if (vid < (unsigned)MAXV)
        keyOfVid[vid] = (unsigned)slab * (unsigned)SLABVOX + (unsigned)(w * 32 + bit);
      ++r;
    }
  }

  // (b) per-point voxel id + top-5-min-index insertion
  unsigned pb = ptBase[slab], pn = ptCnt[slab];
  for (unsigned i = threadIdx.x; i < pn; i += 256) {
    __builtin_prefetch(&recs[(size_t)pb + i + 256], 0, 0);
    unsigned long long rec = recs[(size_t)pb + i];
    unsigned lk = (unsigned)(rec & 0xFFFFu);
    unsigned idx = (unsigned)(rec >> 32);
    unsigned w = lk >> 5, bit = lk & 31u;
    unsigned rank = wp[w] + (unsigned)__popc(bm[w] & ((1u << bit) - 1u));
    unsigned vid = vb + rank;
    if (vid < (unsigned)MAXV) {
      atomicAdd(&cntV[vid], 1u);
      unsigned val = idx;
#pragma unroll
      for (int j = 0; j < MAXP; ++j) {
        unsigned old = atomicMin(&cand[vid * MAXP + j], val);
        if (old == 0xFFFFFFFFu) break;       // filled an empty slot -> done
        val = (old > val) ? old : val;       // displaced larger value cascades
      }
    }
  }
}

// ---------------------------------------------------------------------------
// K7: finalize outputs (coors decode, npts = min(count,5), gather point rows)
// ---------------------------------------------------------------------------
__global__ __launch_bounds__(256) void k_final(const float4* __restrict__ pts4,
                                               const unsigned* __restrict__ keyOfVid,
                                               const unsigned* __restrict__ cntV,
                                               const unsigned* __restrict__ cand,
                                               const unsigned* __restrict__ totalOcc,
                                               float* __restrict__ vox,
                                               int* __restrict__ coors,
                                               int* __restrict__ npts) {
  unsigned vid = blockIdx.x * 256 + threadIdx.x;
  unsigned vnum = *totalOcc;
  if (vnum > (unsigned)MAXV) vnum = (unsigned)MAXV;
  if (vid >= vnum) return;  // stays zero (init)
  unsigned key = keyOfVid[vid];
  unsigned c0 = key / (unsigned)SLABVOX;
  unsigned rem = key % (unsigned)SLABVOX;
  unsigned c1 = rem / (unsigned)GZ;
  unsigned c2 = rem % (unsigned)GZ;
  coors[vid * 3 + 0] = (int)c0;
  coors[vid * 3 + 1] = (int)c1;
  coors[vid * 3 + 2] = (int)c2;
  unsigned np = cntV[vid];
  if (np > (unsigned)MAXP) np = (unsigned)MAXP;
  npts[vid] = (int)np;
  float4* dst = (float4*)(vox + (size_t)vid * MAXP * 4);
  for (unsigned j = 0; j < np; ++j) {
    unsigned p = cand[vid * MAXP + j];
    dst[j] = pts4[p];
  }
}

// ---------------------------------------------------------------------------
extern "C" void kernel_launch(void* const* d_in, const int* in_sizes, int n_in,
                              void* d_out, int out_size, void* d_ws, size_t ws_size,
                              hipStream_t stream) {
  const float4* pts4 = (const float4*)d_in[0];
  const int N = in_sizes[0] / 4;

  // Workspace layout (records first for 8B alignment). ~44 MB for N = 4M.
  char* ws = (char*)d_ws;
  size_t off = 0;
  unsigned long long* recs = (unsigned long long*)(ws);
  off += (size_t)N * 8;
  auto alloc_u32 = [&](size_t n) {
    unsigned* p = (unsigned*)(ws + off);
    off += n * 4;
    return p;
  };
  unsigned* slabCnt = alloc_u32(NSLAB);
  unsigned* slabPtBase = alloc_u32(NSLAB);
  unsigned* slabFill = alloc_u32(NSLAB);
  unsigned* slabOcc = alloc_u32(NSLAB);
  unsigned* slabVoxBase = alloc_u32(NSLAB);
  unsigned* totalOcc = alloc_u32(1);
  unsigned* keyOfVid = alloc_u32(MAXV);
  unsigned* cntV = alloc_u32(MAXV);
  unsigned* cand = alloc_u32((size_t)MAXV * MAXP);
  unsigned* bitmaps = alloc_u32((size_t)NSLAB * NWORDS);

  float* vox = (float*)d_out;
  int* coors = (int*)d_out + (size_t)MAXV * MAXP * 4;
  int* npts = coors + (size_t)MAXV * 3;
  int* vnum = npts + MAXV;
  const int outN = MAXV * MAXP * 4 + MAXV * 3 + MAXV + 1;  // 480001

  k_init<<<(outN + 255) / 256, 256, 0, stream>>>((unsigned*)d_out, outN, cand, cntV,
                                                 keyOfVid, slabCnt, slabFill, slabOcc);
  k_count<<<(N + 4095) / 4096, 256, 0, stream>>>(pts4, N, slabCnt);
  k_exscan<<<1, 1024, 0, stream>>>(slabCnt, slabPtBase, NSLAB, (unsigned*)nullptr,
                                   (int*)nullptr);
  k_scatter<<<(N + 255) / 256, 256, 0, stream>>>(pts4, N, slabPtBase, slabFill, recs);
  k_bitmap<<<NSLAB, 256, 0, stream>>>(recs, slabPtBase, slabCnt, bitmaps, slabOcc);
  k_exscan<<<1, 1024, 0, stream>>>(slabOcc, slabVoxBase, NSLAB, totalOcc, vnum);
  k_assign<<<NSLAB, 256, 0, stream>>>(recs, slabPtBase, slabCnt, bitmaps, slabVoxBase,
                                      keyOfVid, cntV, cand);
  k_final<<<(MAXV + 255) / 256, 256, 0, stream>>>(pts4, keyOfVid, cntV, cand, totalOcc,
                                                  vox, coors, npts);
}